// GCN_17386027614490
// MI455X (gfx1250) — compile-verified
//
#include <hip/hip_runtime.h>

typedef __attribute__((ext_vector_type(2))) float v2f;
typedef __attribute__((ext_vector_type(8))) float v8f;

#define LDS_PAD 132  // 128 + 4 pad: A-frag ds_load_b64 hits all 64 banks conflict-free

// ---------------------------------------------------------------------------
// GEMM: H[M,Nout] = X[M,K] @ W[K,Nout], f32, exact (V_WMMA_F32_16X16X4_F32).
// One block per 16-row slab of X (M = 16 * gridDim.x). Wave w computes the
// 16x16 tile at columns [16w, 16w+16). A-slab staged in LDS, reused by all
// waves; W streamed from global (L2-resident, 64KB).
// ---------------------------------------------------------------------------
__global__ void gcn_gemm_wmma(const float* __restrict__ X,
                              const float* __restrict__ W,
                              float* __restrict__ H,
                              int K, int Nout) {
  __shared__ float As[16 * LDS_PAD];
  const int tid  = threadIdx.x;
  const int wave = tid >> 5;
  const int lane = tid & 31;
  const long tileM = (long)blockIdx.x * 16;

  // Cooperative stage of A slab: 16 x K floats
  for (int i = tid; i < 16 * K; i += blockDim.x) {
    const int r = i / K;
    const int c = i - r * K;
    As[r * LDS_PAD + c] = X[(tileM + r) * (long)K + c];
  }
  __syncthreads();

  const int tileN = wave << 4;
  const int mn = lane & 15;          // M index for A-frag, N index for B-frag
  const int kp = (lane >> 4) << 1;   // lanes 0-15 -> K {0,1}; lanes 16-31 -> K {2,3}

  v8f acc = {};
  for (int k = 0; k < K; k += 4) {
    v2f a, b;
    // A 16x4 f32 fragment (ISA layout): VGPR0=K even, VGPR1=K odd
    a.x = As[mn * LDS_PAD + k + kp];
    a.y = As[mn * LDS_PAD + k + kp + 1];
    // B 4x16 f32 fragment (mirror layout): row = K, striped over lanes by N
    b.x = W[(long)(k + kp) * Nout + tileN + mn];
    b.y = W[(long)(k + kp + 1) * Nout + tileN + mn];
    acc = __builtin_amdgcn_wmma_f32_16x16x4_f32(false, a, false, b,
                                                (short)0, acc, false, false);
  }

  // D layout: VGPR v holds row v (lanes 0-15) / row v+8 (lanes 16-31), col = lane%16
  const long row0 = tileM + ((lane >> 4) << 3);
  const int  col  = tileN + mn;
#pragma unroll
  for (int v = 0; v < 8; ++v)
    H[(row0 + v) * (long)Nout + col] = acc[v];
}

// ---------------------------------------------------------------------------
// Degree / normalization
// ---------------------------------------------------------------------------
__global__ void deg_init(float* __restrict__ deg, int n) {
  int i = blockIdx.x * blockDim.x + threadIdx.x;
  if (i < n) deg[i] = 1.0f;  // self-loop contribution
}

__global__ void deg_count(const long long* __restrict__ ei, float* __restrict__ deg,
                          int E) {
  int e = blockIdx.x * blockDim.x + threadIdx.x;
  if (e < E) {
    int dst = (int)ei[(long)E + e];
    atomicAdd(&deg[dst], 1.0f);
  }
}

__global__ void deg_rsqrt(float* __restrict__ deg, int n) {
  int i = blockIdx.x * blockDim.x + threadIdx.x;
  if (i < n) deg[i] = rsqrtf(deg[i]);  // deg >= 1 always (self-loop)
}

// ---------------------------------------------------------------------------
// agg[i,:] = dinv[i]^2 * H[i,:]   (self-loop term, also zero-initializes agg)
// ---------------------------------------------------------------------------
__global__ void agg_init(const float* __restrict__ H, const float* __restrict__ dinv,
                         float* __restrict__ agg, long total, int C) {
  long idx = (long)blockIdx.x * blockDim.x + threadIdx.x;
  if (idx < total) {
    int i = (int)(idx / C);
    float d = dinv[i];
    agg[idx] = H[idx] * (d * d);
  }
}

// ---------------------------------------------------------------------------
// Edge scatter: agg[dst,:] += dinv[src]*dinv[dst] * H[src,:]
// One thread per (edge, 4 channels); float4 gather + 4 f32 atomics.
// H and agg are L2-resident (<= 51 MB each vs 192 MB L2).
// ---------------------------------------------------------------------------
__global__ void edge_scatter(const long long* __restrict__ ei,
                             const float* __restrict__ H,
                             const float* __restrict__ dinv,
                             float* __restrict__ agg,
                             int E, int C) {
  const int c4 = C >> 2;
  long idx = (long)blockIdx.x * blockDim.x + threadIdx.x;
  long total = (long)E * c4;
  if (idx >= total) return;
  long e  = idx / c4;
  int  cg = (int)(idx - e * c4) << 2;
  int src = (int)ei[e];
  int dst = (int)ei[(long)E + e];
  float w = dinv[src] * dinv[dst];
  const float4 hv = *(const float4*)(H + (long)src * C + cg);
  float* o = agg + (long)dst * C + cg;
  atomicAdd(o + 0, hv.x * w);
  atomicAdd(o + 1, hv.y * w);
  atomicAdd(o + 2, hv.z * w);
  atomicAdd(o + 3, hv.w * w);
}

// ---------------------------------------------------------------------------
// out = relu(agg + bias)   (in-place safe)
// ---------------------------------------------------------------------------
__global__ void bias_relu(const float* __restrict__ agg, const float* __restrict__ b,
                          float* __restrict__ out, long total, int C) {
  long idx = (long)blockIdx.x * blockDim.x + threadIdx.x;
  if (idx < total) {
    int c = (int)(idx % C);
    float v = agg[idx] + b[c];
    out[idx] = v > 0.0f ? v : 0.0f;
  }
}

// ---------------------------------------------------------------------------

static inline unsigned blks(long total, int bs) { return (unsigned)((total + bs - 1) / bs); }

extern "C" void kernel_launch(void* const* d_in, const int* in_sizes, int n_in,
                              void* d_out, int out_size, void* d_ws, size_t ws_size,
                              hipStream_t stream) {
  const float*     x   = (const float*)d_in[0];
  const long long* ei  = (const long long*)d_in[1];   // int64 [2, E]
  const float*     W1  = (const float*)d_in[2];
  const float*     b1  = (const float*)d_in[3];
  const float*     W2  = (const float*)d_in[4];
  const float*     b2  = (const float*)d_in[5];
  float*           out = (float*)d_out;

  const int H1 = in_sizes[3];            // 128
  const int H2 = in_sizes[5];            // 64
  const int IN = in_sizes[2] / H1;       // 128
  const int N  = in_sizes[0] / IN;       // 100000 (= 16 * 6250)
  const int E  = in_sizes[1] / 2;        // 1,600,000

  // Scratch layout: dinv[N] | Hbuf[N*H1] | agg[N*H1]
  char* w = (char*)d_ws;
  size_t off0 = ((size_t)N * 4 + 255) & ~(size_t)255;
  float* dinv = (float*)w;
  float* Hbuf = (float*)(w + off0);
  float* agg  = Hbuf + (size_t)N * H1;

  // --- symmetric normalization: dinv = rsqrt(in-degree incl. self-loop) ---
  deg_init <<<blks(N, 256), 256, 0, stream>>>(dinv, N);
  deg_count<<<blks(E, 256), 256, 0, stream>>>(ei, dinv, E);
  deg_rsqrt<<<blks(N, 256), 256, 0, stream>>>(dinv, N);

  // --- layer 1: h1 = relu(Ahat @ (x @ W1) + b1) ---
  gcn_gemm_wmma<<<N / 16, 32 * (H1 / 16), 0, stream>>>(x, W1, Hbuf, IN, H1);
  {
    long t = (long)N * H1;
    agg_init    <<<blks(t, 256), 256, 0, stream>>>(Hbuf, dinv, agg, t, H1);
    edge_scatter<<<blks((long)E * (H1 / 4), 256), 256, 0, stream>>>(ei, Hbuf, dinv, agg, E, H1);
    bias_relu   <<<blks(t, 256), 256, 0, stream>>>(agg, b1, agg, t, H1);  // h1 = agg
  }

  // --- layer 2: out = relu(Ahat @ (h1 @ W2) + b2), aggregated into d_out ---
  gcn_gemm_wmma<<<N / 16, 32 * (H2 / 16), 0, stream>>>(agg, W2, Hbuf, H1, H2);
  {
    long t = (long)N * H2;
    agg_init    <<<blks(t, 256), 256, 0, stream>>>(Hbuf, dinv, out, t, H2);
    edge_scatter<<<blks((long)E * (H2 / 4), 256), 256, 0, stream>>>(ei, Hbuf, dinv, out, E, H2);
    bias_relu   <<<blks(t, 256), 256, 0, stream>>>(out, b2, out, t, H2);
  }
}